// MultiSimilarityLossSM_88880053223606
// MI455X (gfx1250) — compile-verified
//
#include <hip/hip_runtime.h>
#include <cstdint>
#include <cstddef>

// MultiSimilarityLoss, B = 8192.
// One workgroup per row: row read once from HBM into VGPRs (memory-bound floor
// ~11us at 23.3 TB/s for the 256MB matrix). Labels staged to LDS via CDNA5
// async global->LDS copies (ASYNCcnt), same-label mask kept as a 32-bit
// bitmask per lane, three register-resident passes, wave32 shuffle reductions.

#define MS_BLOCK 256
#define MS_WAVES (MS_BLOCK / 32)
#define MS_B 8192

__device__ __forceinline__ float wredMax(float v) {
#pragma unroll
  for (int m = 16; m > 0; m >>= 1) v = fmaxf(v, __shfl_xor(v, m, 32));
  return v;
}
__device__ __forceinline__ float wredMin(float v) {
#pragma unroll
  for (int m = 16; m > 0; m >>= 1) v = fminf(v, __shfl_xor(v, m, 32));
  return v;
}
__device__ __forceinline__ float wredSum(float v) {
#pragma unroll
  for (int m = 16; m > 0; m >>= 1) v += __shfl_xor(v, m, 32);
  return v;
}

__global__ __launch_bounds__(MS_BLOCK) void ms_row_kernel(
    const float* __restrict__ sim, const int* __restrict__ labels,
    float* __restrict__ row_loss) {
  __shared__ __align__(16) int sh_labels[MS_B];   // 32 KB
  __shared__ float sA[MS_WAVES];
  __shared__ float sB[MS_WAVES];

  const int t = threadIdx.x;
  const int row = blockIdx.x;
  const int wave = t >> 5;
  const int lane = t & 31;

  // ---- stage all 8192 labels into LDS with async copies (ASYNCcnt path),
  //      overlapped with the row's register loads below.
#if defined(__AMDGCN__)
  {
    uint32_t lds_base = (uint32_t)(size_t)(&sh_labels[0]);
#pragma unroll
    for (int i = 0; i < 8; ++i) {
      uint32_t off = (uint32_t)(i * 4096 + t * 16);   // 16B per lane per iter
      uint32_t la = lds_base + off;
      asm volatile("global_load_async_to_lds_b128 %0, %1, %2"
                   :: "v"(la), "v"(off), "s"(labels)
                   : "memory");
    }
  }
#else
  for (int i = t; i < MS_B; i += MS_BLOCK) sh_labels[i] = labels[i];
#endif

  // ---- row -> VGPRs: 32 floats per lane, 8 x b128 coalesced loads
  const float4* rowp =
      reinterpret_cast<const float4*>(sim + (size_t)row * (size_t)MS_B);
  float v[32];
#pragma unroll
  for (int i = 0; i < 8; ++i) {
    float4 q = rowp[i * MS_BLOCK + t];
    v[4 * i + 0] = q.x;
    v[4 * i + 1] = q.y;
    v[4 * i + 2] = q.z;
    v[4 * i + 3] = q.w;
  }

#if defined(__AMDGCN__)
  asm volatile("s_wait_asynccnt 0x0" ::: "memory");
#endif
  __syncthreads();

  const int my_label = sh_labels[row];

  // ---- pass 1: full-row max + same-label bitmask (1 bit per owned element)
  uint32_t same = 0u;
  float lmax = -1e30f;
#pragma unroll
  for (int i = 0; i < 8; ++i) {
    int4 L = *reinterpret_cast<const int4*>(&sh_labels[i * 1024 + 4 * t]);
    if (L.x == my_label) same |= 1u << (4 * i + 0);
    if (L.y == my_label) same |= 1u << (4 * i + 1);
    if (L.z == my_label) same |= 1u << (4 * i + 2);
    if (L.w == my_label) same |= 1u << (4 * i + 3);
#pragma unroll
    for (int c = 0; c < 4; ++c) lmax = fmaxf(lmax, v[4 * i + c]);
  }
  lmax = wredMax(lmax);
  if (lane == 0) sA[wave] = lmax;
  __syncthreads();
  float row_max = sA[0];
#pragma unroll
  for (int w = 1; w < MS_WAVES; ++w) row_max = fmaxf(row_max, sA[w]);
  const float cut = row_max - 1e-5f;   // EPS
  __syncthreads();

  // ---- pass 2: hardest positive (min over pos_mask), hardest negative (max)
  float lminp = 1e30f, lmaxn = -1e30f;
#pragma unroll
  for (int k = 0; k < 32; ++k) {
    const float s = v[k];
    const bool sm = (same >> k) & 1u;
    const float pcand = (sm && (s < cut)) ? s : 1e30f;
    const float ncand = sm ? -1e30f : s;
    lminp = fminf(lminp, pcand);
    lmaxn = fmaxf(lmaxn, ncand);
  }
  lminp = wredMin(lminp);
  lmaxn = wredMax(lmaxn);
  if (lane == 0) { sA[wave] = lminp; sB[wave] = lmaxn; }
  __syncthreads();
  float min_pos = sA[0], max_neg = sB[0];
#pragma unroll
  for (int w = 1; w < MS_WAVES; ++w) {
    min_pos = fminf(min_pos, sA[w]);
    max_neg = fmaxf(max_neg, sB[w]);
  }
  __syncthreads();

  // ---- pass 3: mined soft sums (hardware exp via __expf -> v_exp_f32)
  float psum = 0.f, nsum = 0.f;
#pragma unroll
  for (int k = 0; k < 32; ++k) {
    const float s = v[k];
    const bool sm = (same >> k) & 1u;
    if (sm) {
      if ((s < cut) && (s - 0.1f < max_neg))       // pos_sel
        psum += __expf(-2.0f * (s - 0.5f));
    } else {
      if (s + 0.1f > min_pos)                      // neg_sel
        nsum += __expf(40.0f * (s - 0.5f));
    }
  }
  psum = wredSum(psum);
  nsum = wredSum(nsum);
  if (lane == 0) { sA[wave] = psum; sB[wave] = nsum; }
  __syncthreads();
  if (t == 0) {
    float ps = 0.f, ns = 0.f;
#pragma unroll
    for (int w = 0; w < MS_WAVES; ++w) { ps += sA[w]; ns += sB[w]; }
    // exp terms are strictly positive, so sum>0 <=> any selected
    const float pl = (ps > 0.f) ? log1pf(ps) * (1.0f / 2.0f) : 0.f;
    const float nl = (ns > 0.f) ? log1pf(ns) * (1.0f / 40.0f) : 0.f;
    row_loss[row] = (my_label != 0) ? (pl + nl) : 0.f;
  }
}

__global__ __launch_bounds__(MS_BLOCK) void ms_final_reduce(
    const float* __restrict__ row_loss, float* __restrict__ out, int B) {
  __shared__ float sA[MS_WAVES];
  float s = 0.f;
  for (int i = threadIdx.x; i < B; i += MS_BLOCK) s += row_loss[i];
  s = wredSum(s);
  const int wave = threadIdx.x >> 5;
  const int lane = threadIdx.x & 31;
  if (lane == 0) sA[wave] = s;
  __syncthreads();
  if (threadIdx.x == 0) {
    float tot = 0.f;
#pragma unroll
    for (int w = 0; w < MS_WAVES; ++w) tot += sA[w];
    out[0] = tot / (float)B;
  }
}

extern "C" void kernel_launch(void* const* d_in, const int* in_sizes, int n_in,
                              void* d_out, int out_size, void* d_ws,
                              size_t ws_size, hipStream_t stream) {
  (void)n_in; (void)out_size; (void)ws_size;
  const float* sim = (const float*)d_in[0];
  const int* labels = (const int*)d_in[1];
  float* out = (float*)d_out;
  float* row_loss = (float*)d_ws;          // B floats of scratch
  const int B = in_sizes[1];               // 8192

  ms_row_kernel<<<B, MS_BLOCK, 0, stream>>>(sim, labels, row_loss);
  ms_final_reduce<<<1, MS_BLOCK, 0, stream>>>(row_loss, out, B);
}